// GCN_37606733644135
// MI455X (gfx1250) — compile-verified
//
#include <hip/hip_runtime.h>
#include <hip/hip_bf16.h>

typedef __attribute__((ext_vector_type(2))) float v2f;
typedef __attribute__((ext_vector_type(8))) float v8f;

#define F_IN  128
#define F_H   64
#define F_OUT 16

// ---------------------------------------------------------------------------
// Broadcast a bias vector into an [N, f] accumulator (f is a power of two).
// Folds the "+ b" of each GraphConv layer into the scatter-add accumulator.
// ---------------------------------------------------------------------------
__global__ void gcn_init_bias(float* __restrict__ buf,
                              const float* __restrict__ bias,
                              int total, int fmask) {
  int i = blockIdx.x * blockDim.x + threadIdx.x;
  if (i < total) buf[i] = bias[i & fmask];
}

// ---------------------------------------------------------------------------
// WMMA fp32 GEMM: H[N, Nf] = act(X[N, K]) @ W[K, Nf]
// One wave computes one 16x16 output tile using V_WMMA_F32_16X16X4_F32.
// A layout (16x4 f32): lanes 0-15 hold M=0..15 {K=0,K=1}; lanes 16-31 {K=2,K=3}
// B layout (4x16 f32): lanes 0-15 hold N=0..15 row K=0/K=1; lanes 16-31 K=2/K=3
// C/D (16x16 f32, 8 VGPRs): vgpr v -> M=v (lanes 0-15) / M=v+8 (lanes 16-31)
// ---------------------------------------------------------------------------
template <int K, int NF, bool RELU>
__global__ __launch_bounds__(128) void gcn_gemm_wmma(
    const float* __restrict__ X, const float* __restrict__ W,
    float* __restrict__ H, int nTilesM) {
  const int wave = threadIdx.x >> 5;
  const int lane = threadIdx.x & 31;
  const int gw   = blockIdx.x * (blockDim.x >> 5) + wave;

  constexpr int NT = NF / 16;          // N-tiles per row block
  const int mTile = gw / NT;
  const int nTile = gw % NT;
  if (mTile >= nTilesM) return;        // wave-uniform: EXEC stays all-ones

  const int half = lane >> 4;          // 0: lanes 0-15, 1: lanes 16-31
  const int l    = lane & 15;
  const int row  = mTile * 16 + l;     // A-matrix row owned by this lane
  const int col  = nTile * 16 + l;     // B-matrix column owned by this lane

  const float* __restrict__ xr = X + (size_t)row * K;

  v8f acc = {};
#pragma unroll 4
  for (int k = 0; k < K; k += 4) {
    const int ka = k + 2 * half;       // this lane's K pair (A and B alike)
    v2f a, b;
    a.x = xr[ka + 0];
    a.y = xr[ka + 1];
    if (RELU) {                        // fold relu(agg1 + b1) into the A load
      a.x = fmaxf(a.x, 0.0f);
      a.y = fmaxf(a.y, 0.0f);
    }
    b.x = W[(ka + 0) * NF + col];
    b.y = W[(ka + 1) * NF + col];
    acc = __builtin_amdgcn_wmma_f32_16x16x4_f32(
        /*neg_a=*/false, a, /*neg_b=*/false, b,
        /*c_mod=*/(short)0, acc, /*reuse_a=*/false, /*reuse_b=*/false);
  }

  float* __restrict__ out = H + (size_t)(mTile * 16 + half * 8) * NF + col;
#pragma unroll
  for (int v = 0; v < 8; ++v) out[(size_t)v * NF] = acc[v];
}

// Scalar tail for row counts not divisible by 16 (generic fallback path).
template <int K, int NF, bool RELU>
__global__ void gcn_gemm_tail(const float* __restrict__ X,
                              const float* __restrict__ W,
                              float* __restrict__ H,
                              int rowStart, int rows) {
  int t = blockIdx.x * blockDim.x + threadIdx.x;
  if (t >= rows * NF) return;
  int r = rowStart + t / NF;
  int c = t % NF;
  float s = 0.0f;
  for (int k = 0; k < K; ++k) {
    float x = X[(size_t)r * K + k];
    if (RELU) x = fmaxf(x, 0.0f);
    s += x * W[k * NF + c];
  }
  H[(size_t)r * NF + c] = s;
}

// ---------------------------------------------------------------------------
// Edge aggregation: agg[dst[e], f] += H[src[e], f] * ew[e]
// FPE feature-lanes per edge; contiguous FPE*4-byte gathers (coalesced),
// native fp32 global atomics scatter — accumulator lives in the 192MB L2.
// ---------------------------------------------------------------------------
template <int FPE>
__global__ __launch_bounds__(256) void gcn_edge_scatter(
    const float* __restrict__ H, const int* __restrict__ src,
    const int* __restrict__ dst, const float* __restrict__ ew,
    float* __restrict__ agg, int E) {
  int t = blockIdx.x * blockDim.x + threadIdx.x;
  int e = t / FPE;
  int f = t % FPE;
  if (e >= E) return;
  const float w = ew[e];
  const float v = H[(size_t)src[e] * FPE + f] * w;
  atomicAdd(&agg[(size_t)dst[e] * FPE + f], v);
}

// ---------------------------------------------------------------------------
extern "C" void kernel_launch(void* const* d_in, const int* in_sizes, int n_in,
                              void* d_out, int out_size, void* d_ws, size_t ws_size,
                              hipStream_t stream) {
  const float* X   = (const float*)d_in[0];   // [N, 128]
  const int*   src = (const int*)  d_in[1];   // [E]
  const int*   dst = (const int*)  d_in[2];   // [E]
  const float* ew  = (const float*)d_in[3];   // [E]
  const float* W1  = (const float*)d_in[4];   // [128, 64]
  const float* b1  = (const float*)d_in[5];   // [64]
  const float* W2  = (const float*)d_in[6];   // [64, 16]
  const float* b2  = (const float*)d_in[7];   // [16]
  float* out = (float*)d_out;                 // [N, 16]

  const int N = in_sizes[0] / F_IN;
  const int E = in_sizes[1];

  // Workspace layout: h1 [N,64] | agg1 [N,64] | h2 [N,16]
  float* h1   = (float*)d_ws;
  float* agg1 = h1   + (size_t)N * F_H;
  float* h2   = agg1 + (size_t)N * F_H;

  const int nTilesM = N / 16;
  const int tailRows = N - nTilesM * 16;

  // ---- Layer 1 ----
  {   // agg1 <- b1  (fold bias into accumulator)
    int total = N * F_H;
    gcn_init_bias<<<(total + 255) / 256, 256, 0, stream>>>(agg1, b1, total, F_H - 1);
  }
  {   // h1 <- X @ W1  (WMMA f32)
    int waves = nTilesM * (F_H / 16);
    gcn_gemm_wmma<F_IN, F_H, false><<<(waves + 3) / 4, 128, 0, stream>>>(X, W1, h1, nTilesM);
    if (tailRows > 0) {
      int tt = tailRows * F_H;
      gcn_gemm_tail<F_IN, F_H, false><<<(tt + 255) / 256, 256, 0, stream>>>(
          X, W1, h1, nTilesM * 16, tailRows);
    }
  }
  {   // agg1 += scatter(h1[src] * ew -> dst)
    long long total = (long long)E * F_H;
    gcn_edge_scatter<F_H><<<(unsigned)((total + 255) / 256), 256, 0, stream>>>(
        h1, src, dst, ew, agg1, E);
  }

  // ---- Layer 2 ----
  {   // out <- b2
    int total = N * F_OUT;
    gcn_init_bias<<<(total + 255) / 256, 256, 0, stream>>>(out, b2, total, F_OUT - 1);
  }
  {   // h2 <- relu(agg1) @ W2  (WMMA f32, relu folded into A loads)
    int waves = nTilesM * (F_OUT / 16);
    gcn_gemm_wmma<F_H, F_OUT, true><<<(waves + 3) / 4, 128, 0, stream>>>(agg1, W2, h2, nTilesM);
    if (tailRows > 0) {
      int tt = tailRows * F_OUT;
      gcn_gemm_tail<F_H, F_OUT, true><<<(tt + 255) / 256, 256, 0, stream>>>(
          agg1, W2, h2, nTilesM * 16, tailRows);
    }
  }
  {   // out += scatter(h2[src] * ew -> dst)
    long long total = (long long)E * F_OUT;
    gcn_edge_scatter<F_OUT><<<(unsigned)((total + 255) / 256), 256, 0, stream>>>(
        h2, src, dst, ew, out, E);
  }
}